// GraphEncoder_42013370089719
// MI455X (gfx1250) — compile-verified
//
#include <hip/hip_runtime.h>

// ---------------- problem constants (match reference) ----------------
#define NN   100000     // nodes
#define E2   1600000    // directed edges (2*E)
#define DV   128
#define DE   16
#define DH   128
#define DO   128
#define GB   64         // graphs in batch

typedef __attribute__((ext_vector_type(2))) float v2f;
typedef __attribute__((ext_vector_type(8))) float v8f;

// One K=4 step of f32 WMMA: D(16x16) += A(16x4) * B(4x16)
__device__ __forceinline__ v8f wmma_k4(v8f c, float a0, float a1, float b0, float b1) {
    v2f a = {a0, a1};
    v2f b = {b0, b1};
    return __builtin_amdgcn_wmma_f32_16x16x4_f32(
        /*neg_a=*/false, a, /*neg_b=*/false, b,
        /*c_mod=*/(short)0, c, /*reuse_a=*/false, /*reuse_b=*/false);
}

// ---------------- init: deg=1 (self loop), agg_e=1 (self-loop attr), pool/cnt=0
__global__ void k_init(float* __restrict__ deg, float* __restrict__ agge,
                       float* __restrict__ pool, float* __restrict__ cnt) {
    int i = blockIdx.x * blockDim.x + threadIdx.x;
    if (i < NN)      deg[i]  = 1.0f;
    if (i < NN * DE) agge[i] = 1.0f;
    if (i < GB * DH) pool[i] = 0.0f;
    if (i < GB)      cnt[i]  = 0.0f;
}

// ---------------- degree: deg[dst] += 1 per directed edge
__global__ void k_deg(const int* __restrict__ dst, float* __restrict__ deg) {
    int i = blockIdx.x * blockDim.x + threadIdx.x;
    if (i < E2) atomicAdd(&deg[dst[i]], 1.0f);
}

// ---------------- folded small weights: Wle2 = W_le @ W_lin[128:], cvec = b_le@W_lin[128:] + b_lin
__global__ void k_smallw(const float* __restrict__ W_le, const float* __restrict__ b_le,
                         const float* __restrict__ W_lin, const float* __restrict__ b_lin,
                         float* __restrict__ Wle2, float* __restrict__ cvec) {
    int i = blockIdx.x * blockDim.x + threadIdx.x;
    if (i < DE * DH) {
        int d = i >> 7, n = i & 127;
        float s = 0.f;
        for (int k = 0; k < DH; ++k)
            s += W_le[d * DH + k] * W_lin[(DH + k) * DH + n];
        Wle2[i] = s;
    } else if (i < DE * DH + DH) {
        int n = i - DE * DH;
        float s = b_lin[n];
        for (int k = 0; k < DH; ++k)
            s += b_le[k] * W_lin[(DH + k) * DH + n];
        cvec[n] = s;
    }
}

// ---------------- dinv = rsqrt(deg); y = dinv*x; agg_y = y (self loop). Wave per node, float4/lane.
__global__ void k_scale(const float* __restrict__ x, const float* __restrict__ deg,
                        float* __restrict__ dinv, float* __restrict__ y,
                        float* __restrict__ aggy) {
    int t = blockIdx.x * blockDim.x + threadIdx.x;
    int node = t >> 5, lane = t & 31;
    if (node >= NN) return;
    float d = rsqrtf(deg[node]);
    if (lane == 0) dinv[node] = d;
    float4 xv = ((const float4*)x)[node * 32 + lane];
    float4 yv;
    yv.x = d * xv.x; yv.y = d * xv.y; yv.z = d * xv.z; yv.w = d * xv.w;
    ((float4*)y)[node * 32 + lane]    = yv;
    ((float4*)aggy)[node * 32 + lane] = yv;
}

// ---------------- 128-wide gather/scatter-add: agg[dst] += feat[src], wave per edge
__global__ void k_scatter128(const float* __restrict__ feat, float* __restrict__ agg,
                             const int* __restrict__ src, const int* __restrict__ dst) {
    int t = blockIdx.x * blockDim.x + threadIdx.x;
    int e = t >> 5, lane = t & 31;
    if (e >= E2) return;
    int s = src[e], d = dst[e];
    float4 v = ((const float4*)feat)[s * 32 + lane];
    float* base = agg + d * DH + lane * 4;
    atomicAdd(base + 0, v.x);
    atomicAdd(base + 1, v.y);
    atomicAdd(base + 2, v.z);
    atomicAdd(base + 3, v.w);
}

// ---------------- edge-attr scatter: agg_e[dst] += edge_attr[e>>1] (repeat-interleaved)
__global__ void k_scatter_e(const float* __restrict__ ea, const int* __restrict__ dst,
                            float* __restrict__ agge) {
    int t = blockIdx.x * blockDim.x + threadIdx.x;
    int e = t >> 4, c = t & 15;
    if (e >= E2) return;
    atomicAdd(&agge[dst[e] * DE + c], ea[(e >> 1) * DE + c]);
}

// ---------------- copy N*128 floats (h -> agg_h self-loop init)
__global__ void k_copy(const float* __restrict__ a, float* __restrict__ b) {
    int t = blockIdx.x * blockDim.x + threadIdx.x;
    if (t < NN * 32) ((float4*)b)[t] = ((const float4*)a)[t];
}

#define LDSW 132   // padded A-tile stride (floats): 132 mod 64 = 4 -> conflict-free

// ---------------- GEMM1: h = relu(dinv ⊙ (agg_y @ W_gcn) + b_gcn). 16 rows/block, 8 waves.
__global__ __launch_bounds__(256) void k_gemm1(const float* __restrict__ aggy,
                                               const float* __restrict__ Wg,
                                               const float* __restrict__ bias,
                                               const float* __restrict__ dinv,
                                               float* __restrict__ h) {
    __shared__ float As[16 * LDSW];
    int rowbase = blockIdx.x * 16;
    int t = threadIdx.x;
    for (int i = t; i < 16 * 32; i += 256) {            // 512 float4 loads
        int r = i >> 5, c4 = i & 31;
        ((float4*)(As + r * LDSW))[c4] = ((const float4*)(aggy + (rowbase + r) * DH))[c4];
    }
    __syncthreads();
    int wave = t >> 5, lane = t & 31, half = lane >> 4, l = lane & 15;
    int col = wave * 16 + l;
    v8f c = {};
    for (int kk = 0; kk < 32; ++kk) {
        int kb = kk * 4 + 2 * half;                     // ISA A/B f32 fragment layout
        c = wmma_k4(c, As[l * LDSW + kb], As[l * LDSW + kb + 1],
                       Wg[kb * DH + col], Wg[(kb + 1) * DH + col]);
    }
    float bc = bias[col];
    for (int r = 0; r < 8; ++r) {
        int row = rowbase + r + 8 * half;               // C/D layout: VGPR r -> rows r / r+8
        float v = dinv[row] * c[r] + bc;
        h[row * DH + col] = fmaxf(v, 0.f);
    }
}

// ---------------- GEMM2: h2 = relu(agg_h @ W1 + agg_e @ Wle2 + deg ⊗ cvec)
#define ESW 20     // padded 16x16 tile stride
__global__ __launch_bounds__(256) void k_gemm2(const float* __restrict__ aggh,
                                               const float* __restrict__ agge,
                                               const float* __restrict__ W1,
                                               const float* __restrict__ Wle2,
                                               const float* __restrict__ cvec,
                                               const float* __restrict__ deg,
                                               float* __restrict__ h2) {
    __shared__ float As[16 * LDSW];
    __shared__ float Es[16 * ESW];
    int rowbase = blockIdx.x * 16;
    int t = threadIdx.x;
    for (int i = t; i < 16 * 32; i += 256) {
        int r = i >> 5, c4 = i & 31;
        ((float4*)(As + r * LDSW))[c4] = ((const float4*)(aggh + (rowbase + r) * DH))[c4];
    }
    if (t < 64) {
        int r = t >> 2, c4 = t & 3;
        ((float4*)(Es + r * ESW))[c4] = ((const float4*)(agge + (rowbase + r) * DE))[c4];
    }
    __syncthreads();
    int wave = t >> 5, lane = t & 31, half = lane >> 4, l = lane & 15;
    int col = wave * 16 + l;
    v8f c = {};
    for (int kk = 0; kk < 32; ++kk) {                   // K=128 node-feature part
        int kb = kk * 4 + 2 * half;
        c = wmma_k4(c, As[l * LDSW + kb], As[l * LDSW + kb + 1],
                       W1[kb * DH + col], W1[(kb + 1) * DH + col]);
    }
    for (int kk = 0; kk < 4; ++kk) {                    // K=16 edge-attr part
        int kb = kk * 4 + 2 * half;
        c = wmma_k4(c, Es[l * ESW + kb], Es[l * ESW + kb + 1],
                       Wle2[kb * DH + col], Wle2[(kb + 1) * DH + col]);
    }
    float cc = cvec[col];
    for (int r = 0; r < 8; ++r) {
        int row = rowbase + r + 8 * half;
        float v = c[r] + deg[row] * cc;
        h2[row * DH + col] = fmaxf(v, 0.f);
    }
}

// ---------------- pooling scatter: pool[batch[v]] += h2[v]; cnt[batch[v]] += 1
__global__ void k_pool(const float* __restrict__ h2, const int* __restrict__ batch,
                       float* __restrict__ pool, float* __restrict__ cnt) {
    int t = blockIdx.x * blockDim.x + threadIdx.x;
    int node = t >> 5, lane = t & 31;
    if (node >= NN) return;
    int g = batch[node];
    float4 v = ((const float4*)h2)[node * 32 + lane];
    float* base = pool + g * DH + lane * 4;
    atomicAdd(base + 0, v.x);
    atomicAdd(base + 1, v.y);
    atomicAdd(base + 2, v.z);
    atomicAdd(base + 3, v.w);
    if (lane == 0) atomicAdd(&cnt[g], 1.0f);
}

// ---------------- final: out = (pool/cnt) @ W_out + b_out  (64x128 x 128x128)
__global__ __launch_bounds__(256) void k_final(const float* __restrict__ pool,
                                               const float* __restrict__ cnt,
                                               const float* __restrict__ Wo,
                                               const float* __restrict__ bo,
                                               float* __restrict__ out) {
    __shared__ float As[16 * LDSW];
    int rowbase = blockIdx.x * 16;
    int t = threadIdx.x;
    for (int i = t; i < 16 * 128; i += 256) {
        int r = i >> 7, cc = i & 127;
        As[r * LDSW + cc] = pool[(rowbase + r) * DH + cc] / fmaxf(cnt[rowbase + r], 1.0f);
    }
    __syncthreads();
    int wave = t >> 5, lane = t & 31, half = lane >> 4, l = lane & 15;
    int col = wave * 16 + l;
    v8f c = {};
    for (int kk = 0; kk < 32; ++kk) {
        int kb = kk * 4 + 2 * half;
        c = wmma_k4(c, As[l * LDSW + kb], As[l * LDSW + kb + 1],
                       Wo[kb * DO + col], Wo[(kb + 1) * DO + col]);
    }
    float bc = bo[col];
    for (int r = 0; r < 8; ++r) {
        int row = rowbase + r + 8 * half;
        out[row * DO + col] = c[r] + bc;
    }
}

extern "C" void kernel_launch(void* const* d_in, const int* in_sizes, int n_in,
                              void* d_out, int out_size, void* d_ws, size_t ws_size,
                              hipStream_t stream) {
    const float* x     = (const float*)d_in[0];
    const int*   eidx  = (const int*)  d_in[1];
    const float* eattr = (const float*)d_in[2];
    const int*   batch = (const int*)  d_in[3];
    const float* W_gcn = (const float*)d_in[4];
    const float* b_gcn = (const float*)d_in[5];
    const float* W_le  = (const float*)d_in[6];
    const float* b_le  = (const float*)d_in[7];
    const float* W_lin = (const float*)d_in[8];
    const float* b_lin = (const float*)d_in[9];
    const float* W_out = (const float*)d_in[10];
    const float* b_out = (const float*)d_in[11];
    float* out = (float*)d_out;

    const int* src = eidx;          // edge_index[0]
    const int* dst = eidx + E2;     // edge_index[1]

    // workspace carve-up (all 16B aligned)
    float* ws   = (float*)d_ws;
    float* deg  = ws;               // N
    float* dinv = deg  + NN;        // N
    float* agge = dinv + NN;        // N*16
    float* Wle2 = agge + NN * DE;   // 2048
    float* cvec = Wle2 + DE * DH;   // 128
    float* pool = cvec + DH;        // 64*128
    float* cnt  = pool + GB * DH;   // 64
    float* bufA = cnt  + GB;        // N*128  (y -> h -> h2)
    float* bufB = bufA + NN * DH;   // N*128  (agg_y -> agg_h)

    const int B = 256;
    k_init     <<<(NN * DE + B - 1) / B, B, 0, stream>>>(deg, agge, pool, cnt);
    k_deg      <<<(E2 + B - 1) / B,      B, 0, stream>>>(dst, deg);
    k_smallw   <<<(DE * DH + DH + B - 1) / B, B, 0, stream>>>(W_le, b_le, W_lin, b_lin, Wle2, cvec);
    k_scale    <<<(NN * 32 + B - 1) / B, B, 0, stream>>>(x, deg, dinv, bufA, bufB);
    k_scatter128<<<(E2 * 32) / B,        B, 0, stream>>>(bufA, bufB, src, dst);   // agg_y
    k_gemm1    <<<NN / 16,               B, 0, stream>>>(bufB, W_gcn, b_gcn, dinv, bufA); // h
    k_scatter_e<<<(E2 * 16) / B,         B, 0, stream>>>(eattr, dst, agge);
    k_copy     <<<(NN * 32 + B - 1) / B, B, 0, stream>>>(bufA, bufB);             // agg_h = h
    k_scatter128<<<(E2 * 32) / B,        B, 0, stream>>>(bufA, bufB, src, dst);   // agg_h += h[src]
    k_gemm2    <<<NN / 16,               B, 0, stream>>>(bufB, agge, W_lin, Wle2, cvec, deg, bufA); // h2
    k_pool     <<<(NN * 32 + B - 1) / B, B, 0, stream>>>(bufA, batch, pool, cnt);
    k_final    <<<GB / 16,               B, 0, stream>>>(pool, cnt, W_out, b_out, out);
}